// EarthAttention3D_39238821216902
// MI455X (gfx1250) — compile-verified
//
#include <hip/hip_runtime.h>
#include <hip/hip_bf16.h>

// EarthAttention3D fully fused for MI455X (gfx1250):
//   one workgroup per (l, t) window; QKV proj + biased softmax attention +
//   output projection all in LDS (~258 KB of the 320 KB/WGP budget).
//   All matrix math via V_WMMA_F32_16X16X4_F32 (exact fp32 tensor path).
//   All WMMA operands are contiguous 8-byte LDS pairs -> ds_load_(2addr_)b64,
//   conflict-free bank patterns. Softmax runs 2 lanes per row with shfl_xor
//   pair reductions so all 288 threads stay busy.

#define DIM   192
#define HEADS 6
#define HD    32
#define NWIN  144
#define TW    64

// LDS strides (floats); chosen so lane-stride mod 64 banks is conflict-free
#define QS_LD 36    // Q/K tiles: 144 x 32          (36*l mod 64 -> distinct)
#define VT_LD 148   // V transposed: 32 x 144       (20*l mod 64 -> distinct)
#define PS_LD 148   // score matrix: 144 x 144
#define OS_LD 196   // attention output: 144 x 192  (4*l mod 64 -> distinct)
#define WT_LD 196   // transposed W1/W2 tile: 96 x 192

typedef float v2f __attribute__((ext_vector_type(2)));
typedef float v8f __attribute__((ext_vector_type(8)));

__device__ __forceinline__ v8f wmma_f32_16x16x4(v2f a, v2f b, v8f c) {
#if defined(__gfx1250__) && __has_builtin(__builtin_amdgcn_wmma_f32_16x16x4_f32)
  return __builtin_amdgcn_wmma_f32_16x16x4_f32(false, a, false, b, (short)0, c,
                                               false, false);
#else
  asm volatile("v_wmma_f32_16x16x4_f32 %0, %1, %2, %0"
               : "+v"(c)
               : "v"(a), "v"(b));
  return c;
#endif
}

__device__ __forceinline__ v2f lds2(const float* p) {
  return *(const v2f*)p;  // contiguous 8B LDS pair -> ds_load_b64
}

__launch_bounds__(288, 1)
__global__ void earth_attn_fused(const float* __restrict__ x,
                                 const float* __restrict__ mask,
                                 const float* __restrict__ W1,
                                 const float* __restrict__ b1,
                                 const float* __restrict__ W2,
                                 const float* __restrict__ b2,
                                 const float* __restrict__ bias_table,
                                 float* __restrict__ out) {
  __shared__ float Qs[NWIN * QS_LD];   // 144 x 32 (padded)
  __shared__ float Ks[NWIN * QS_LD];   // 144 x 32 (padded)
  __shared__ float Vt[HD * VT_LD];     // V transposed: 32 x 144 (padded)
  __shared__ float PW[NWIN * PS_LD];   // scores <-> transposed weight tile
  __shared__ float Os[NWIN * OS_LD];   // attention output 144 x 192 (padded)

  const int tid  = threadIdx.x;
  const int wave = tid >> 5;
  const int lane = tid & 31;
  const int ln   = lane & 15;
  const int hi   = lane >> 4;          // half-wave: K offset 0/2, M offset 0/8
  const int bid  = blockIdx.x;         // l*64 + t
  const int l    = bid >> 6;
  const int t    = bid & 63;
  const int r0   = bid * NWIN;

  const float scale = 0.17677669529663687f;  // 32^-0.5
  const int mrow = wave * 16 + ln;           // A-operand row owned by this lane

  // Preload this wave's 16x192 x-slice as WMMA A-operand registers (reused by
  // all 6 heads' QKV GEMMs -> zero global traffic in the inner loops).
  v2f xa[48];
#pragma unroll
  for (int i = 0; i < 48; ++i) {
    xa[i] = *(const v2f*)&x[(size_t)(r0 + mrow) * DIM + i * 4 + hi * 2];
  }

  // Precompute (z, h, w) decomposition of the 9 key tokens this lane touches
  // in the bias epilogue: m = nt*16 + ln.  pos_index(n, m) =
  //   (z_n + 2*z_m)*828 + (h_n + 6*h_m)*23 + (w_n - w_m + 11)
  int mz[9], mh[9], mw[9];
#pragma unroll
  for (int nt = 0; nt < 9; ++nt) {
    int m = nt * 16 + ln;
    int z = m / 72, r = m - z * 72, hh = r / 12;
    mz[nt] = z;
    mh[nt] = hh;
    mw[nt] = r - hh * 12;
  }

  for (int h = 0; h < HEADS; ++h) {
    // ---- Phase 1: stage W1 head slice TRANSPOSED (96 cols x 192 k) into PW --
    for (int i = tid; i < 192 * 24; i += 288) {
      int k  = i / 24;
      int c4 = i - k * 24;             // float4 chunk within 96 local cols
      int seg = c4 >> 3;               // 0=q, 1=k, 2=v
      int gcol = seg * DIM + h * HD + (c4 & 7) * 4;
      float4 v = *(const float4*)&W1[k * (3 * DIM) + gcol];
      PW[(c4 * 4 + 0) * WT_LD + k] = v.x;
      PW[(c4 * 4 + 1) * WT_LD + k] = v.y;
      PW[(c4 * 4 + 2) * WT_LD + k] = v.z;
      PW[(c4 * 4 + 3) * WT_LD + k] = v.w;
    }
    __syncthreads();

    // ---- Phase 2: QKV GEMM  C(144x96) = x(144x192) @ W1h + b1h ----
    v8f acc[6];
#pragma unroll
    for (int nt = 0; nt < 6; ++nt) {
      int c   = nt * 16 + ln;
      int seg = c >> 5;
      float bv = b1[seg * DIM + h * HD + (c & 31)];
#pragma unroll
      for (int j = 0; j < 8; ++j) acc[nt][j] = bv;
    }
    for (int kk = 0; kk < 48; ++kk) {
      v2f a = xa[kk];
      int ko = kk * 4 + hi * 2;
#pragma unroll
      for (int nt = 0; nt < 6; ++nt) {
        v2f b = lds2(&PW[(nt * 16 + ln) * WT_LD + ko]);
        acc[nt] = wmma_f32_16x16x4(a, b, acc[nt]);
      }
    }
    // Scatter C tile: Q (pre-scaled) and K row-major, V transposed
#pragma unroll
    for (int nt = 0; nt < 6; ++nt) {
      int c = nt * 16 + ln;
#pragma unroll
      for (int j = 0; j < 8; ++j) {
        int n = wave * 16 + j + hi * 8;
        float v = acc[nt][j];
        if (c < 32)      Qs[n * QS_LD + c]          = v * scale;
        else if (c < 64) Ks[n * QS_LD + (c - 32)]   = v;
        else             Vt[(c - 64) * VT_LD + n]   = v;
      }
    }
    __syncthreads();  // QKV visible; all waves done reading PW (W1 slice)

    // ---- Phase 3: S = Q @ K^T  (144x144, K=32) + bias + mask -> PW ----
    v8f s[9];
#pragma unroll
    for (int nt = 0; nt < 9; ++nt)
#pragma unroll
      for (int j = 0; j < 8; ++j) s[nt][j] = 0.0f;
    for (int kk = 0; kk < 8; ++kk) {
      int ko = kk * 4 + hi * 2;
      v2f a = lds2(&Qs[mrow * QS_LD + ko]);
#pragma unroll
      for (int nt = 0; nt < 9; ++nt) {
        v2f b = lds2(&Ks[(nt * 16 + ln) * QS_LD + ko]);  // B = K^T
        s[nt] = wmma_f32_16x16x4(a, b, s[nt]);
      }
    }
    {
      const float* mbase = mask + (size_t)l * NWIN * NWIN;
      const float* btab  = bias_table + t * HEADS + h;   // + idx*64*6
#pragma unroll
      for (int j = 0; j < 8; ++j) {
        int n  = wave * 16 + j + hi * 8;                 // query token
        int zn = n / 72, rn = n - zn * 72, hn = rn / 12;
        int wn = rn - hn * 12;
        const float* mrow_p = mbase + n * NWIN;
        float* prow = &PW[n * PS_LD];
#pragma unroll
        for (int nt = 0; nt < 9; ++nt) {
          int m = nt * 16 + ln;                          // key token
          int idx = (zn + 2 * mz[nt]) * 828 + (hn + 6 * mh[nt]) * 23 +
                    (wn - mw[nt] + 11);
          float bias = btab[(size_t)idx * (TW * HEADS)];
          prow[m] = s[nt][j] + bias + mrow_p[m];
        }
      }
    }
    __syncthreads();

    // ---- Phase 4: row softmax, 2 lanes per row (288 = 2*144) ----
    {
      int row  = tid >> 1;
      int halfo = (tid & 1) * 72;
      float* prow = &PW[row * PS_LD];
      float mx = -3.4e38f;
      for (int m = halfo; m < halfo + 72; ++m) mx = fmaxf(mx, prow[m]);
      mx = fmaxf(mx, __shfl_xor(mx, 1, 32));
      float sum = 0.0f;
      for (int m = halfo; m < halfo + 72; ++m) {
        float e = __expf(prow[m] - mx);
        prow[m] = e;
        sum += e;
      }
      sum += __shfl_xor(sum, 1, 32);
      float inv = 1.0f / sum;
      for (int m = halfo; m < halfo + 72; ++m) prow[m] *= inv;
    }
    __syncthreads();

    // ---- Phase 5: O_h = P @ V  (144x32, K=144) -> Os cols [h*32, h*32+32) --
    v8f o[2];
#pragma unroll
    for (int nt = 0; nt < 2; ++nt)
#pragma unroll
      for (int j = 0; j < 8; ++j) o[nt][j] = 0.0f;
    for (int kk = 0; kk < 36; ++kk) {
      int ko = kk * 4 + hi * 2;
      v2f a = lds2(&PW[mrow * PS_LD + ko]);
#pragma unroll
      for (int nt = 0; nt < 2; ++nt) {
        v2f b = lds2(&Vt[(nt * 16 + ln) * VT_LD + ko]);  // B[k=m][d] = Vt[d][m]
        o[nt] = wmma_f32_16x16x4(a, b, o[nt]);
      }
    }
#pragma unroll
    for (int nt = 0; nt < 2; ++nt)
#pragma unroll
      for (int j = 0; j < 8; ++j) {
        int n = wave * 16 + j + hi * 8;
        Os[n * OS_LD + h * HD + nt * 16 + ln] = o[nt][j];
      }
    __syncthreads();  // PW (scores) free for next head's W1 slice
  }

  // ---- Phase 6: out = Os(144x192) @ W2 + b2, two 96-column halves ----
  for (int half = 0; half < 2; ++half) {
    // stage W2 columns [half*96, half*96+96) transposed into PW
    for (int i = tid; i < 192 * 24; i += 288) {
      int k  = i / 24;
      int c4 = i - k * 24;
      float4 v = *(const float4*)&W2[k * DIM + half * 96 + c4 * 4];
      PW[(c4 * 4 + 0) * WT_LD + k] = v.x;
      PW[(c4 * 4 + 1) * WT_LD + k] = v.y;
      PW[(c4 * 4 + 2) * WT_LD + k] = v.z;
      PW[(c4 * 4 + 3) * WT_LD + k] = v.w;
    }
    __syncthreads();
    v8f acc[6];
#pragma unroll
    for (int nt = 0; nt < 6; ++nt) {
      float bv = b2[half * 96 + nt * 16 + ln];
#pragma unroll
      for (int j = 0; j < 8; ++j) acc[nt][j] = bv;
    }
    for (int kk = 0; kk < 48; ++kk) {
      int ko = kk * 4 + hi * 2;
      v2f a = lds2(&Os[mrow * OS_LD + ko]);
#pragma unroll
      for (int nt = 0; nt < 6; ++nt) {
        v2f b = lds2(&PW[(nt * 16 + ln) * WT_LD + ko]);
        acc[nt] = wmma_f32_16x16x4(a, b, acc[nt]);
      }
    }
#pragma unroll
    for (int nt = 0; nt < 6; ++nt)
#pragma unroll
      for (int j = 0; j < 8; ++j) {
        int n = wave * 16 + j + hi * 8;
        out[(size_t)(r0 + n) * DIM + half * 96 + nt * 16 + ln] = acc[nt][j];
      }
    __syncthreads();
  }
}

extern "C" void kernel_launch(void* const* d_in, const int* in_sizes, int n_in,
                              void* d_out, int out_size, void* d_ws, size_t ws_size,
                              hipStream_t stream) {
  (void)in_sizes; (void)n_in; (void)out_size; (void)d_ws; (void)ws_size;
  const float* x          = (const float*)d_in[0];
  const float* mask       = (const float*)d_in[1];
  const float* W1         = (const float*)d_in[2];
  const float* b1         = (const float*)d_in[3];
  const float* W2         = (const float*)d_in[4];
  const float* b2         = (const float*)d_in[5];
  const float* bias_table = (const float*)d_in[6];
  float* out = (float*)d_out;

  // 8 (L) * 64 (T) = 512 windows; 9 waves (288 threads) per window.
  earth_attn_fused<<<dim3(512), dim3(288), 0, stream>>>(
      x, mask, W1, b1, W2, b2, bias_table, out);
}